// SpGraphAttentionLayer_20014547599820
// MI455X (gfx1250) — compile-verified
//
#include <hip/hip_runtime.h>
#include <hip/hip_bf16.h>

typedef __attribute__((ext_vector_type(2))) float v2f;
typedef __attribute__((ext_vector_type(8))) float v8f;

#define GAT_B 4
#define GAT_N 20000
#define GAT_M (GAT_B * GAT_N)        // 80000 rows
#define GAT_INF 256
#define GAT_OUTF 64
#define GAT_ALPHA 0.2f

// ---------------------------------------------------------------------------
// Kernel 1: h = x @ W  (M=80000, K=256, N=64) with V_WMMA_F32_16X16X4_F32,
// fused per-row scores s_src = h . a1, s_dst = h . a2.
// Block = 128 threads = 4 waves; each wave owns a 16-row x 64-col tile.
// W is staged in LDS pair-interleaved: Wp[k/2][n] = (W[k][n], W[k+1][n]),
// so every B fragment is a single aligned ds_load_b64 into an even VGPR pair.
// ---------------------------------------------------------------------------
__global__ void __launch_bounds__(128)
gat_gemm_kernel(const float* __restrict__ x, const float* __restrict__ W,
                const float* __restrict__ a, float* __restrict__ h,
                float* __restrict__ s_src, float* __restrict__ s_dst)
{
    __shared__ float Wlds[GAT_INF * GAT_OUTF];   // 64 KB, fits WGP LDS

    const int tid = threadIdx.x;
    // cooperative W load (coalesced float4 reads, pair-interleaved LDS stores)
    {
        const float4* Wg = (const float4*)W;
        #pragma unroll
        for (int it = 0; it < (GAT_INF * GAT_OUTF / 4) / 128; ++it) {
            const int i = tid + it * 128;        // float4 index
            const float4 w4 = Wg[i];
            const int k   = i >> 4;              // source row (16 float4s per row)
            const int c   = (i & 15) * 4;        // source col base
            const int kp  = k >> 1;
            const int klo = k & 1;
            float* dst = &Wlds[kp * (2 * GAT_OUTF) + klo];
            dst[(c + 0) * 2] = w4.x;
            dst[(c + 1) * 2] = w4.y;
            dst[(c + 2) * 2] = w4.z;
            dst[(c + 3) * 2] = w4.w;
        }
    }
    __syncthreads();

    const int lane = tid & 31;
    const int wave = tid >> 5;
    const int rowBase = (blockIdx.x * 4 + wave) * 16;   // 1250 blocks * 4 waves * 16 = 80000
    const int n  = lane & 15;            // column within 16-wide tile / A row
    const int kk = (lane >> 4) * 2;      // K sub-offset for A/B fragments
    const int m0 = (lane < 16) ? 0 : 8;  // C/D row-half offset

    v8f c0 = {}, c1 = {}, c2 = {}, c3 = {};

    // A fragments: one global_load_b64 per k-step
    const v2f* xp = (const v2f*)(x + (size_t)(rowBase + n) * GAT_INF);
    // B fragments: one ds_load_b64 each
    const v2f* Wp = (const v2f*)Wlds;

    #pragma unroll 4
    for (int k0 = 0; k0 < GAT_INF; k0 += 4) {
        const int kp = (k0 + kk) >> 1;
        const v2f af = xp[kp];
        const v2f* wr = Wp + kp * GAT_OUTF + n;
        const v2f b0 = wr[0];
        const v2f b1 = wr[16];
        const v2f b2 = wr[32];
        const v2f b3 = wr[48];

        c0 = __builtin_amdgcn_wmma_f32_16x16x4_f32(false, af, false, b0, (short)0, c0, false, false);
        c1 = __builtin_amdgcn_wmma_f32_16x16x4_f32(false, af, false, b1, (short)0, c1, false, false);
        c2 = __builtin_amdgcn_wmma_f32_16x16x4_f32(false, af, false, b2, (short)0, c2, false, false);
        c3 = __builtin_amdgcn_wmma_f32_16x16x4_f32(false, af, false, b3, (short)0, c3, false, false);
    }

    // per-lane attention vector slices: a1[16t+n], a2[16t+n]
    float a1v[4], a2v[4];
    #pragma unroll
    for (int t = 0; t < 4; ++t) {
        a1v[t] = a[t * 16 + n];
        a2v[t] = a[GAT_OUTF + t * 16 + n];
    }

    #pragma unroll
    for (int r = 0; r < 8; ++r) {
        const int row = rowBase + m0 + r;
        const float v0 = c0[r], v1 = c1[r], v2 = c2[r], v3 = c3[r];
        float* hr = h + (size_t)row * GAT_OUTF + n;
        hr[0]  = v0;
        hr[16] = v1;
        hr[32] = v2;
        hr[48] = v3;
        float ps = v0 * a1v[0] + v1 * a1v[1] + v2 * a1v[2] + v3 * a1v[3];
        float pd = v0 * a2v[0] + v1 * a2v[1] + v2 * a2v[2] + v3 * a2v[3];
        // reduce over the 16 columns (xor masks < 16 stay within each half-wave)
        #pragma unroll
        for (int off = 1; off < 16; off <<= 1) {
            ps += __shfl_xor(ps, off, 32);
            pd += __shfl_xor(pd, off, 32);
        }
        if (n == 0) {
            s_src[row] = ps;
            s_dst[row] = pd;
        }
    }
}

// ---------------------------------------------------------------------------
// Kernel 2: per-edge softmax weights + scatter-add.
// 64 threads per edge -> coalesced 64-float gather of h and 64 coalesced
// global_atomic_add_f32 into num; lane 0 adds edge_e into rowsum.
// h (20.5 MB) and num (20.5 MB) both live in the 192 MB L2.
// ---------------------------------------------------------------------------
__global__ void __launch_bounds__(256)
gat_edge_kernel(const int* __restrict__ edge_b, const int* __restrict__ edge_src,
                const int* __restrict__ edge_dst,
                const float* __restrict__ s_src, const float* __restrict__ s_dst,
                const float* __restrict__ h,
                float* __restrict__ num, float* __restrict__ rowsum, int E)
{
    const int e = blockIdx.x * 4 + (threadIdx.x >> 6);
    if (e >= E) return;
    const int lane64 = threadIdx.x & 63;

    const int b = edge_b[e];
    const int s = edge_src[e];
    const int d = edge_dst[e];
    const int srcIdx = b * GAT_N + s;
    const int dstIdx = b * GAT_N + d;

    const float score = s_src[srcIdx] + s_dst[dstIdx];
    const float lrelu = (score > 0.0f) ? score : GAT_ALPHA * score;
    const float ev = expf(-lrelu);

    const float w = ev * h[(size_t)dstIdx * GAT_OUTF + lane64];
    atomicAdd(&num[(size_t)srcIdx * GAT_OUTF + lane64], w);
    if (lane64 == 0)
        atomicAdd(&rowsum[srcIdx], ev);
}

// ---------------------------------------------------------------------------
// Kernel 3: h_prime = num / rowsum ; NaN (0/0 for empty segments) -> 0 ; ELU.
// ---------------------------------------------------------------------------
__global__ void __launch_bounds__(256)
gat_final_kernel(const float* __restrict__ num, const float* __restrict__ rowsum,
                 float* __restrict__ out, int total)
{
    const int i = blockIdx.x * 256 + threadIdx.x;
    if (i >= total) return;
    const float r = rowsum[i >> 6];
    float v = num[i] / r;
    if (v != v) v = 0.0f;               // NaN -> 0 (matches jnp.where(isnan,...))
    out[i] = (v > 0.0f) ? v : expm1f(v); // elu, alpha=1
}

// ---------------------------------------------------------------------------
// Launch wrapper. Workspace layout (floats):
//   [0, 5.12M)        h
//   [5.12M, 10.24M)   num
//   [10.24M, ...)     s_src(80000) | s_dst(80000) | rowsum(80000)
// ---------------------------------------------------------------------------
extern "C" void kernel_launch(void* const* d_in, const int* in_sizes, int n_in,
                              void* d_out, int out_size, void* d_ws, size_t ws_size,
                              hipStream_t stream)
{
    const float* x = (const float*)d_in[0];
    const float* W = (const float*)d_in[1];
    const float* a = (const float*)d_in[2];
    const int* edge_b   = (const int*)d_in[3];
    const int* edge_src = (const int*)d_in[4];
    const int* edge_dst = (const int*)d_in[5];
    const int E = in_sizes[3];

    const size_t HN = (size_t)GAT_M * GAT_OUTF;   // 5,120,000
    float* ws     = (float*)d_ws;
    float* h      = ws;
    float* num    = ws + HN;
    float* s_src  = ws + 2 * HN;
    float* s_dst  = s_src + GAT_M;
    float* rowsum = s_dst + GAT_M;

    hipMemsetAsync(num, 0, HN * sizeof(float), stream);
    hipMemsetAsync(rowsum, 0, (size_t)GAT_M * sizeof(float), stream);

    // 80000 rows / (4 waves * 16 rows) = 1250 blocks
    gat_gemm_kernel<<<GAT_M / 64, 128, 0, stream>>>(x, W, a, h, s_src, s_dst);

    gat_edge_kernel<<<(E + 3) / 4, 256, 0, stream>>>(edge_b, edge_src, edge_dst,
                                                     s_src, s_dst, h, num, rowsum, E);

    gat_final_kernel<<<((int)HN + 255) / 256, 256, 0, stream>>>(num, rowsum,
                                                                (float*)d_out, (int)HN);
}